// SelfAttention_38070590112267
// MI455X (gfx1250) — compile-verified
//
#include <hip/hip_runtime.h>
#include <hip/hip_bf16.h>

// ---------------------------------------------------------------------------
// Problem constants (match reference)
// ---------------------------------------------------------------------------
#define TT    2048
#define DIN   1024
#define DOUT  1024
#define NHEAD 8
#define DH    128   // DOUT / NHEAD

typedef __attribute__((ext_vector_type(2))) float v2f;
typedef __attribute__((ext_vector_type(8))) float v8f;

// ---------------------------------------------------------------------------
// Generic fp32 WMMA GEMM:  C[M,N] = A[M,K] * B[K,N]
//   TRANSB=true : B[k][n] = W[n*ldw + k]   (W row-major [N,K], i.e. C = A*W^T)
//   TRANSB=false: B[k][n] = W[k*ldw + n]   (W row-major [K,N])
// Block = 256 threads = 8 waves. Each wave owns a 16x64 strip:
//   rows  m0 = (blockIdx.x*8 + wave)*16
//   cols  n0 = blockIdx.y*64  (4 WMMA tiles, A-fragment reused 4x)
// V_WMMA_F32_16X16X4_F32 fragment layout (ISA 7.12.2):
//   A 16x4 : lane<16 -> {K=k,k+1}, lane>=16 -> {K=k+2,k+3}; row = m0 + (lane&15)
//   B 4x16 : mirrored (lane<16 rows k,k+1; lane>=16 rows k+2,k+3); col = n0+(lane&15)
//   C 16x16: VGPR r -> row m0 + r + 8*(lane>>4), col n0 + (lane&15)
// All M,N used here are multiples of 128/64 and K of 4 -> no tail handling.
// ---------------------------------------------------------------------------
template <bool TRANSB>
__global__ __launch_bounds__(256) void wmma_gemm_f32(
    const float* __restrict__ A, const float* __restrict__ W,
    float* __restrict__ C, int M, int N, int Kdim, int lda, int ldw, int ldc)
{
    const int wave = threadIdx.x >> 5;
    const int lane = threadIdx.x & 31;
    const int half = lane >> 4;      // 0: K pair {0,1}, 1: K pair {2,3}
    const int l16  = lane & 15;

    const int m0 = (blockIdx.x * 8 + wave) * 16;
    const int n0 = blockIdx.y * 64;
    if (m0 >= M || n0 >= N) return;  // wave-uniform; grid sized exactly anyway

    v8f acc0 = {}, acc1 = {}, acc2 = {}, acc3 = {};

    const float* Arow = A + (size_t)(m0 + l16) * lda;

#pragma unroll 2
    for (int k = 0; k < Kdim; k += 4) {
        const int ka = k + 2 * half;
        // A fragment: contiguous pair -> global_load_b64
        v2f a = *(const v2f*)(Arow + ka);

        v2f b0, b1, b2, b3;
        if (TRANSB) {
            // B[k][n] = W[n*ldw + k]: per-lane pair {ka,ka+1} is contiguous
            const float* w = W + (size_t)(n0 + l16) * ldw + ka;
            b0 = *(const v2f*)(w);
            b1 = *(const v2f*)(w + (size_t)16 * ldw);
            b2 = *(const v2f*)(w + (size_t)32 * ldw);
            b3 = *(const v2f*)(w + (size_t)48 * ldw);
        } else {
            // B row-major [K,N]: two b32 loads per tile (rows ka, ka+1)
            const float* w = W + (size_t)ka * ldw + n0 + l16;
            b0.x = w[0];  b0.y = w[ldw];
            b1.x = w[16]; b1.y = w[ldw + 16];
            b2.x = w[32]; b2.y = w[ldw + 32];
            b3.x = w[48]; b3.y = w[ldw + 48];
        }

        acc0 = __builtin_amdgcn_wmma_f32_16x16x4_f32(false, a, false, b0, (short)0, acc0, false, false);
        acc1 = __builtin_amdgcn_wmma_f32_16x16x4_f32(false, a, false, b1, (short)0, acc1, false, false);
        acc2 = __builtin_amdgcn_wmma_f32_16x16x4_f32(false, a, false, b2, (short)0, acc2, false, false);
        acc3 = __builtin_amdgcn_wmma_f32_16x16x4_f32(false, a, false, b3, (short)0, acc3, false, false);
    }

#pragma unroll
    for (int r = 0; r < 8; ++r) {
        float* crow = C + (size_t)(m0 + r + 8 * half) * ldc + l16;
        crow[n0]      = acc0[r];
        crow[n0 + 16] = acc1[r];
        crow[n0 + 32] = acc2[r];
        crow[n0 + 48] = acc3[r];
    }
}

// ---------------------------------------------------------------------------
// Row L2-normalize: xu[t,:] = x[t,:] / max(||x[t,:]||, 1e-12)
// ---------------------------------------------------------------------------
__global__ __launch_bounds__(256) void rownorm_kernel(const float* __restrict__ x,
                                                      float* __restrict__ xu)
{
    __shared__ float red[256];
    const int t = blockIdx.x, tid = threadIdx.x;
    const float* xr = x + (size_t)t * DIN;
    float s = 0.f;
    for (int d = tid; d < DIN; d += 256) { float v = xr[d]; s += v * v; }
    red[tid] = s; __syncthreads();
    for (int o = 128; o > 0; o >>= 1) { if (tid < o) red[tid] += red[tid + o]; __syncthreads(); }
    const float inv = 1.0f / fmaxf(sqrtf(red[0]), 1e-12f);
    float* xo = xu + (size_t)t * DIN;
    for (int d = tid; d < DIN; d += 256) xo[d] = xr[d] * inv;
}

__device__ __forceinline__ float block_reduce_sum(float v, float* red)
{
    const int tid = threadIdx.x;
    red[tid] = v; __syncthreads();
    for (int o = 128; o > 0; o >>= 1) { if (tid < o) red[tid] += red[tid + o]; __syncthreads(); }
    float r = red[0]; __syncthreads();
    return r;
}

// ---------------------------------------------------------------------------
// Per-row: e = energies + AP  -> softmax (in place), raw entropy, dep factor.
//   AP[t, s] : i=s>>1, ang = t / 10000^(2i/DIN); even col sin, odd col cos
//   ent_raw[t] = (log S1 - S2/S1) / log(T)   (S2 = sum (e-m) e^{e-m})
//   dep[t]    = (1 - (sum sim*p)/S1) / (T - sum sim)
// ---------------------------------------------------------------------------
__global__ __launch_bounds__(256) void softmax_stats_kernel(
    float* __restrict__ att, const float* __restrict__ sim,
    float* __restrict__ ent, float* __restrict__ dep)
{
    __shared__ float row[TT];
    __shared__ float red[256];
    const int t = blockIdx.x, tid = threadIdx.x;
    const float LN1E4 = 9.210340371976184f;          // ln(10000)

    float lmax = -3.402823466e38f;
    float* arow = att + (size_t)t * TT;
    for (int s = tid; s < TT; s += 256) {
        const int i = s >> 1;
        const float ang = (float)t * __expf(-(2.0f * (float)i / (float)DIN) * LN1E4);
        const float ap  = (s & 1) ? __cosf(ang) : __sinf(ang);
        const float e   = arow[s] + ap;
        row[s] = e;
        lmax = fmaxf(lmax, e);
    }
    red[tid] = lmax; __syncthreads();
    for (int o = 128; o > 0; o >>= 1) { if (tid < o) red[tid] = fmaxf(red[tid], red[tid + o]); __syncthreads(); }
    const float m = red[0]; __syncthreads();

    const float* srow = sim + (size_t)t * TT;
    float s1 = 0.f, s2 = 0.f, sd = 0.f, ss = 0.f;
    for (int s = tid; s < TT; s += 256) {
        const float e = row[s] - m;
        const float p = __expf(e);
        row[s] = p;
        s1 += p; s2 += e * p;
        const float sv = srow[s];
        sd += sv * p; ss += sv;
    }
    const float S1 = block_reduce_sum(s1, red);
    const float S2 = block_reduce_sum(s2, red);
    const float SD = block_reduce_sum(sd, red);
    const float SS = block_reduce_sum(ss, red);

    const float inv = 1.0f / S1;
    for (int s = tid; s < TT; s += 256) arow[s] = row[s] * inv;

    if (tid == 0) {
        ent[t] = (__logf(S1) - S2 * inv) / __logf((float)TT);
        dep[t] = (1.0f - SD * inv) / ((float)TT - SS);
    }
}

// entropy L1-normalization across all rows of a head: ent /= max(sum|ent|,1e-12)
__global__ __launch_bounds__(256) void ent_norm_kernel(float* __restrict__ ent)
{
    __shared__ float red[256];
    const int tid = threadIdx.x;
    float s = 0.f;
    for (int t = tid; t < TT; t += 256) s += fabsf(ent[t]);
    const float S = block_reduce_sum(s, red);
    const float inv = 1.0f / fmaxf(S, 1e-12f);
    for (int t = tid; t < TT; t += 256) ent[t] *= inv;
}

// att_win[i,j] = att[i,j] * (1 + dep[j])   (in place)
__global__ __launch_bounds__(256) void colscale_kernel(float* __restrict__ att,
                                                       const float* __restrict__ dep)
{
    const size_t idx = (size_t)blockIdx.x * 256 + threadIdx.x;
    const int j = (int)(idx & (TT - 1));
    att[idx] *= (1.0f + dep[j]);
}

// ycat[t, hoff+o] += ent[t]*W2[o,DH] + dep[t]*W2[o,DH+1]
__global__ __launch_bounds__(256) void chars_add_kernel(
    float* __restrict__ ycat, const float* __restrict__ W2,
    const float* __restrict__ ent, const float* __restrict__ dep, int hoff)
{
    const int idx = blockIdx.x * 256 + threadIdx.x;
    const int t = idx / DH, o = idx % DH;
    const float* w = W2 + (size_t)o * (DH + 2) + DH;
    ycat[(size_t)t * DOUT + hoff + o] += ent[t] * w[0] + dep[t] * w[1];
}

// ---------------------------------------------------------------------------
extern "C" void kernel_launch(void* const* d_in, const int* in_sizes, int n_in,
                              void* d_out, int out_size, void* d_ws, size_t ws_size,
                              hipStream_t stream)
{
    const float* x     = (const float*)d_in[0];   // [T, DIN]
    const float* Wq    = (const float*)d_in[1];   // [H, DH, DIN]
    const float* Wk    = (const float*)d_in[2];
    const float* Wv    = (const float*)d_in[3];
    const float* Wout2 = (const float*)d_in[4];   // [DH, DH+2]
    const float* Wout  = (const float*)d_in[5];   // [DIN, DOUT]

    float* out      = (float*)d_out;                  // [T, DIN]
    float* att_last = out + (size_t)TT * DIN;         // [T, T] (head 7 att_win)

    // Workspace layout (~52 MB)
    float* ws   = (float*)d_ws;
    float* xu   = ws;                                 // [T, DIN]
    float* sim  = xu   + (size_t)TT * DIN;            // [T, T]
    float* attw = sim  + (size_t)TT * TT;             // [T, T]  (heads 0..6)
    float* Qb   = attw + (size_t)TT * TT;             // [T, DH]
    float* Kb   = Qb   + (size_t)TT * DH;
    float* Vb   = Kb   + (size_t)TT * DH;
    float* yb   = Vb   + (size_t)TT * DH;             // [T, DH]
    float* ycat = yb   + (size_t)TT * DH;             // [T, DOUT]
    float* ent  = ycat + (size_t)TT * DOUT;           // [T]
    float* dep  = ent  + TT;                          // [T]

    const dim3 blk(256);

    // 1) xu = normalize(x);  sim = xu * xu^T
    rownorm_kernel<<<TT, blk, 0, stream>>>(x, xu);
    wmma_gemm_f32<true><<<dim3(TT / 128, TT / 64), blk, 0, stream>>>(
        xu, xu, sim, TT, TT, DIN, DIN, DIN, TT);

    for (int h = 0; h < NHEAD; ++h) {
        const float* Wqh = Wq + (size_t)h * DH * DIN;
        const float* Wkh = Wk + (size_t)h * DH * DIN;
        const float* Wvh = Wv + (size_t)h * DH * DIN;
        float* attb = (h == NHEAD - 1) ? att_last : attw;

        // 2) Q/K/V = x * W^T   [T, DH]
        wmma_gemm_f32<true><<<dim3(TT / 128, DH / 64), blk, 0, stream>>>(
            x, Wqh, Qb, TT, DH, DIN, DIN, DIN, DH);
        wmma_gemm_f32<true><<<dim3(TT / 128, DH / 64), blk, 0, stream>>>(
            x, Wkh, Kb, TT, DH, DIN, DIN, DIN, DH);
        wmma_gemm_f32<true><<<dim3(TT / 128, DH / 64), blk, 0, stream>>>(
            x, Wvh, Vb, TT, DH, DIN, DIN, DIN, DH);

        // 3) energies = Q * K^T  -> attb
        wmma_gemm_f32<true><<<dim3(TT / 128, TT / 64), blk, 0, stream>>>(
            Qb, Kb, attb, TT, TT, DH, DH, DH, TT);

        // 4) softmax(+AP) in place, entropy + dep stats
        softmax_stats_kernel<<<TT, blk, 0, stream>>>(attb, sim, ent, dep);
        ent_norm_kernel<<<1, blk, 0, stream>>>(ent);

        // 5) att_win = att * (1 + dep[col])  (in place; head 7 lands in d_out)
        colscale_kernel<<<(TT * TT) / 256, blk, 0, stream>>>(attb, dep);

        // 6) y = att_win * V   [T, DH]   (B row-major)
        wmma_gemm_f32<false><<<dim3(TT / 128, DH / 64), blk, 0, stream>>>(
            attb, Vb, yb, TT, DH, TT, TT, DH, DH);

        // 7) y_s = y * Wout2[:, :DH]^T  into ycat head slice, then +ent/dep cols
        wmma_gemm_f32<true><<<dim3(TT / 128, DH / 64), blk, 0, stream>>>(
            yb, Wout2, ycat + (size_t)h * DH, TT, DH, DH, DH, DH + 2, DOUT);
        chars_add_kernel<<<(TT * DH) / 256, blk, 0, stream>>>(ycat, Wout2, ent, dep, h * DH);
    }

    // 8) out = ycat * Wout^T   [T, DIN]
    wmma_gemm_f32<true><<<dim3(TT / 128, DIN / 64), blk, 0, stream>>>(
        ycat, Wout, out, TT, DIN, DOUT, DOUT, DOUT, DIN);
}